// NormalizePixelPts_19318762897442
// MI455X (gfx1250) — compile-verified
//
#include <hip/hip_runtime.h>

// out[f,d] = in[f,d] * scale(d); scale pattern is compile-time known.
// Pure streaming kernel: 548 MB @ 23.3 TB/s => ~23.5 us floor. Bandwidth-bound:
// design goal is saturated clause'd b128 NT vmem; scale is computed in VALU
// (branchless) so zero extra memory traffic is spent on it.

typedef float v4f __attribute__((ext_vector_type(4)));

#define FEAT_DIM   4178u
#define NUM_FRAMES 16384u
#define N4         17117184u      // NUM_FRAMES*(FEAT_DIM+1)/4
#define THREADS    256
#define BLOCKS     16716          // BLOCKS*THREADS*4 == N4 exactly
#define STRIDE4    (BLOCKS * THREADS)

// Branchless scale lookup: compiles to v_cndmask chains, no exec-mask branching.
__device__ __forceinline__ float scale_of(unsigned d) {
    const float INV_W = 1.0f / 1280.0f;
    const float INV_H = 1.0f / 720.0f;
    unsigned r = d % 285u;                       // meaningful only when d < 570
    bool tail = (d >= 570u);                     // joint-offset region
    unsigned p = (tail ? d : r) & 1u;            // parity bit
    bool flip  = (!tail) & (r >= 1u) & (r <= 80u);       // first dist block: odd = x
    bool one   = (!tail) & ((r == 0u) | (r == 81u) | (r >= 164u)); // activations etc.
    float pair = ((p != 0u) != flip) ? INV_H : INV_W;
    return one ? 1.0f : pair;
}

__device__ __forceinline__ v4f scales4(unsigned i4) {
    const float INV_W = 1.0f / 1280.0f;
    const float INV_H = 1.0f / 720.0f;
    unsigned flat = i4 * 4u;          // always even
    unsigned d0   = flat % FEAT_DIM;  // even, since FEAT_DIM is even
    v4f s;
    if (d0 >= 570u && d0 <= 4174u) {
        // fast path (~86% of columns, wave-coherent): static per-slot parity
        s.x = INV_W; s.y = INV_H; s.z = INV_W; s.w = INV_H;
    } else {
        // straight-line branchless path (no divergent control flow inside)
        unsigned d1 = d0 + 1u; if (d1 >= FEAT_DIM) d1 -= FEAT_DIM;
        unsigned d2 = d0 + 2u; if (d2 >= FEAT_DIM) d2 -= FEAT_DIM;
        unsigned d3 = d0 + 3u; if (d3 >= FEAT_DIM) d3 -= FEAT_DIM;
        s.x = scale_of(d0);
        s.y = scale_of(d1);
        s.z = scale_of(d2);
        s.w = scale_of(d3);
    }
    return s;
}

__global__ __launch_bounds__(THREADS)
void normalize_pixel_pts_kernel(const v4f* __restrict__ in, v4f* __restrict__ out) {
    unsigned i0 = blockIdx.x * THREADS + threadIdx.x;
    unsigned i1 = i0 + STRIDE4;
    unsigned i2 = i0 + 2u * STRIDE4;
    unsigned i3 = i0 + 3u * STRIDE4;

    // 4 independent b128 NT loads in flight (single s_clause) before dependent ALU
    v4f a0 = __builtin_nontemporal_load(in + i0);
    v4f a1 = __builtin_nontemporal_load(in + i1);
    v4f a2 = __builtin_nontemporal_load(in + i2);
    v4f a3 = __builtin_nontemporal_load(in + i3);

    // scale computation depends only on indices -> overlaps load latency
    v4f s0 = scales4(i0);
    v4f s1 = scales4(i1);
    v4f s2 = scales4(i2);
    v4f s3 = scales4(i3);

    a0 *= s0;
    a1 *= s1;
    a2 *= s2;
    a3 *= s3;

    __builtin_nontemporal_store(a0, out + i0);
    __builtin_nontemporal_store(a1, out + i1);
    __builtin_nontemporal_store(a2, out + i2);
    __builtin_nontemporal_store(a3, out + i3);
}

extern "C" void kernel_launch(void* const* d_in, const int* in_sizes, int n_in,
                              void* d_out, int out_size, void* d_ws, size_t ws_size,
                              hipStream_t stream) {
    const v4f* in  = (const v4f*)d_in[0];
    v4f*       out = (v4f*)d_out;
    normalize_pixel_pts_kernel<<<BLOCKS, THREADS, 0, stream>>>(in, out);
}